// DetInfer_20126216749159
// MI455X (gfx1250) — compile-verified
//
#include <hip/hip_runtime.h>
#include <stdint.h>

#define HW      65536       // 256*256
#define KTOP    100
#define NTHR_A  1024

// dynamic LDS layout for stage A (total 264208 B < 320 KB WGP LDS)
#define HEAT_BYTES   (HW * 4)            // 262144: staged 256x256 f32 heatmap
#define BINS_OFF     HEAT_BYTES          // 256 u32 histogram bins
#define COMP_OFF     (BINS_OFF + 1024)   // 128 u64 sort keys
#define SCAL_OFF     (COMP_OFF + 1024)   // 4 u32 scalars
#define SMEM_A       (SCAL_OFF + 16)

typedef __attribute__((ext_vector_type(4))) unsigned int u32x4;
typedef __attribute__((ext_vector_type(8))) int i32x8;
typedef __attribute__((ext_vector_type(4))) int i32x4;

__device__ __constant__ float c_dim_mean[4][3] = {
    {0.45f, 1.1f, 0.45f},
    {6.2896f, 37.7375f, 11.0068f},
    {4.7718f, 28.6307f, 9.5436f},
    {3.0893f, 18.5356f, 7.7232f}};
__device__ __constant__ float c_dim_std[4][3] = {
    {0.0f, 0.0f, 0.0f},
    {0.0782f, 0.4693f, 0.1369f},
    {0.016f, 0.0958f, 0.0319f},
    {0.0811f, 0.4868f, 0.2028f}};
__device__ __constant__ float c_alpha[4] = {
    0.0f, 1.57079632679489662f, 3.14159265358979324f, -1.57079632679489662f};

// order-preserving float -> u32 key (larger float => larger key)
__device__ __forceinline__ unsigned f2key(float f) {
  unsigned u = __float_as_uint(f);
  return (u & 0x80000000u) ? ~u : (u | 0x80000000u);
}
__device__ __forceinline__ float key2f(unsigned k) {
  unsigned u = (k & 0x80000000u) ? (k ^ 0x80000000u) : ~k;
  return __uint_as_float(u);
}

// 3x3 NMS-suppressed score at pixel idx, as ordered key (suppressed -> 0.0f)
__device__ __forceinline__ unsigned suppressed_key(const float* heat, int idx) {
  const int y = idx >> 8, x = idx & 255;
  const float v = heat[idx];
  float m = v;
#pragma unroll
  for (int dy = -1; dy <= 1; ++dy) {
    int yy = y + dy;
    if ((unsigned)yy >= 256u) continue;
    const float* row = heat + (yy << 8);
#pragma unroll
    for (int dx = -1; dx <= 1; ++dx) {
      int xx = x + dx;
      if ((unsigned)xx >= 256u) continue;
      m = fmaxf(m, row[xx]);
    }
  }
  float s = (m == v) ? v : 0.0f;
  return f2key(s);
}

__device__ __forceinline__ float wrap_pi(float a) {
  const float PI  = 3.14159265358979323846f;
  const float TPI = 6.28318530717958647692f;
  a = (a > PI)  ? a - TPI : a;
  a = (a < -PI) ? a + TPI : a;
  return a;
}

// ---------------- Stage A: per-(b,c) NMS + exact top-100 ----------------
__global__ __launch_bounds__(NTHR_A) void detinfer_topk_kernel(
    const float* __restrict__ det_cls,
    float* __restrict__ s1val,
    int*   __restrict__ s1ind) {
  const int bc  = blockIdx.x;          // b*4 + c, 64 blocks
  const int tid = threadIdx.x;
  extern __shared__ unsigned char smem[];
  float*    heat = (float*)smem;
  unsigned* bins = (unsigned*)(smem + BINS_OFF);
  unsigned long long* comp = (unsigned long long*)(smem + COMP_OFF);
  unsigned* scal = (unsigned*)(smem + SCAL_OFF);

  const float* src = det_cls + (size_t)bc * HW;

  // ---- DMA the whole 256x256 class map into LDS via the Tensor Data Mover ----
#if defined(__gfx1250__) && __has_builtin(__builtin_amdgcn_tensor_load_to_lds)
  if (tid < 32) {  // one wave issues the descriptor; TDM ignores EXEC
    unsigned long long ga = (unsigned long long)(size_t)src;
    unsigned lds_base = (unsigned)(size_t)heat;  // LDS aperture: low 32 bits = LDS offset
    // D# group0: count=1, lds_addr, global_addr[56:0], type=2
    u32x4 g0 = {1u, lds_base,
                (unsigned)(ga & 0xFFFFFFFFull),
                ((unsigned)((ga >> 32) & 0x01FFFFFFull)) | 0x80000000u};
    // D# group1: data_size=4B; tensor 256x256; tile 256x256; dim0 stride 256
    i32x8 g1 = {0x00020000,
                (int)(256u << 16),   // tensor_dim0 = 256 (bits 79:48, lo16 here)
                (int)(256u << 16),   // tensor_dim1 = 256
                (int)(256u << 16),   // tile_dim0   = 256
                256,                 // tile_dim1   = 256, tile_dim2 = 0
                256,                 // tensor_dim0_stride = 256
                0,
                1};                  // tensor_dim1_stride = 65536 (bits 47:16)
    i32x4 gz4 = {0, 0, 0, 0};
    i32x8 gz8 = {0, 0, 0, 0, 0, 0, 0, 0};
    __builtin_amdgcn_tensor_load_to_lds(g0, g1, gz4, gz4, gz8, 0);
    __builtin_amdgcn_s_wait_tensorcnt(0);
  }
#else
  {
    const float4* s4 = (const float4*)src;
    float4* d4 = (float4*)heat;
    for (int i = tid; i < HW / 4; i += NTHR_A) d4[i] = s4[i];
  }
#endif
  __syncthreads();

  // ---- radix-select the key of the 100th-largest suppressed score ----
  unsigned prefix = 0u, want = KTOP;
  for (int shift = 24; shift >= 0; shift -= 8) {
    for (int i = tid; i < 256; i += NTHR_A) bins[i] = 0u;
    __syncthreads();
    for (int i = tid; i < HW; i += NTHR_A) {
      unsigned k = suppressed_key(heat, i);
      bool match = (shift == 24) || ((k >> ((shift + 8) & 31)) == prefix);
      if (match) atomicAdd(&bins[(k >> shift) & 255u], 1u);
    }
    __syncthreads();
    if (tid == 0) {
      unsigned cum = 0; int d = 255;
      for (; d > 0; --d) {
        if (cum + bins[d] >= want) break;
        cum += bins[d];
      }
      scal[0] = (prefix << 8) | (unsigned)d;
      scal[1] = want - cum;
    }
    __syncthreads();
    prefix = scal[0]; want = scal[1];
    __syncthreads();
  }
  const unsigned T = prefix;       // key of the 100th-largest value
  const unsigned needEq = want;    // how many ties at T to take (smallest idx first)

  // ---- collect survivors ----
  for (int i = tid; i < 128; i += NTHR_A) comp[i] = ~0ull;
  if (tid == 0) scal[2] = 0u;
  __syncthreads();
  for (int i = tid; i < HW; i += NTHR_A) {
    unsigned k = suppressed_key(heat, i);
    if (k > T) {
      unsigned p = atomicAdd(&scal[2], 1u);
      if (p < KTOP) comp[p] = (((unsigned long long)(~k)) << 32) | (unsigned)i;
    }
  }
  __syncthreads();
  const unsigned base = scal[2];   // == KTOP - needEq
  int lastIdx = -1;
  for (unsigned t = 0; t < needEq; ++t) {   // usually exactly 1 pass
    if (tid == 0) scal[3] = 0xFFFFFFFFu;
    __syncthreads();
    for (int i = tid; i < HW; i += NTHR_A)
      if (i > lastIdx && suppressed_key(heat, i) == T) atomicMin(&scal[3], (unsigned)i);
    __syncthreads();
    unsigned m = scal[3];
    if (tid == 0 && base + t < KTOP)
      comp[base + t] = (((unsigned long long)(~T)) << 32) | m;
    lastIdx = (int)m;
    __syncthreads();
  }

  // ---- bitonic sort of 128 composites: ascending (~key, idx) == top_k order ----
  for (int k2 = 2; k2 <= 128; k2 <<= 1) {
    for (int j = k2 >> 1; j > 0; j >>= 1) {
      if (tid < 128) {
        int ixj = tid ^ j;
        if (ixj > tid) {
          unsigned long long a = comp[tid], b2 = comp[ixj];
          bool up = ((tid & k2) == 0);
          if ((a > b2) == up) { comp[tid] = b2; comp[ixj] = a; }
        }
      }
      __syncthreads();
    }
  }

  if (tid < KTOP) {
    unsigned long long c = comp[tid];
    unsigned k   = ~(unsigned)(c >> 32);
    unsigned idx = (unsigned)(c & 0xFFFFFFFFull);
    s1val[bc * KTOP + tid] = key2f(k);
    s1ind[bc * KTOP + tid] = (int)idx;
  }
}

// ---------------- Stage B: per-batch merge + box decode ----------------
__global__ __launch_bounds__(512) void detinfer_decode_kernel(
    const float* __restrict__ det_reg,
    const float* __restrict__ s1val,
    const int*   __restrict__ s1ind,
    float* __restrict__ out) {
  const int b   = blockIdx.x;
  const int tid = threadIdx.x;
  __shared__ unsigned long long comp[512];

  if (tid < 4 * KTOP) {
    float v = s1val[b * 400 + tid];
    unsigned k = f2key(v);
    comp[tid] = (((unsigned long long)(~k)) << 32) | (unsigned)tid; // tie: pos asc
  } else {
    comp[tid] = ~0ull;
  }
  __syncthreads();
  for (int k2 = 2; k2 <= 512; k2 <<= 1) {
    for (int j = k2 >> 1; j > 0; j >>= 1) {
      int ixj = tid ^ j;
      if (ixj > tid) {
        unsigned long long a = comp[tid], b2 = comp[ixj];
        bool up = ((tid & k2) == 0);
        if ((a > b2) == up) { comp[tid] = b2; comp[ixj] = a; }
      }
      __syncthreads();
    }
  }

  if (tid < KTOP) {
    unsigned long long c = comp[tid];
    unsigned key = ~(unsigned)(c >> 32);
    int p = (int)(unsigned)(c & 0xFFFFFFFFull);   // cls*100 + rank
    float score = key2f(key);
    int cls = p / KTOP;
    int ind = s1ind[b * 400 + p];

    // det_reg channel layout: (B, 4*22, H, W) with channel = cls*22 + r
    const float* rbase = det_reg + ((size_t)b * 88 + (size_t)cls * 22) * HW + (size_t)ind;
    float sel[22];
#pragma unroll
    for (int r = 0; r < 22; ++r) sel[r] = rbase[(size_t)r * HW];

    float d0 = sel[3] * c_dim_std[cls][0] + c_dim_mean[cls][0];
    float d1 = sel[4] * c_dim_std[cls][1] + c_dim_mean[cls][1];
    float d2 = sel[5] * c_dim_std[cls][2] + c_dim_mean[cls][2];
    float z  = sel[2] + d0 * 0.5f;
    float xs = (float)(ind & 255) + sel[0];
    float ys = (float)(ind >> 8)  + sel[1];
    float xs_w = xs * (90.0f / 256.0f) - 45.0f;
    float ys_w = (256.0f - ys) * (90.0f / 256.0f);

    // softmax(l0,l1)[1] is monotonic in (l1 - l0): argmax via diffs, first-max ties
    int bidx = 0;
    float best = sel[7] - sel[6];
#pragma unroll
    for (int i = 1; i < 4; ++i) {
      float dlt = sel[6 + 2 * i + 1] - sel[6 + 2 * i];
      if (dlt > best) { best = dlt; bidx = i; }
    }
    float off_s = sel[14 + 2 * bidx];
    float off_c = sel[15 + 2 * bidx];
    float alpha = atan2f(off_s, off_c) + c_alpha[bidx];
    float ray   = atan2f(xs_w, ys_w);
    float roty  = wrap_pi(alpha + ray);
    alpha = wrap_pi(alpha);

    const int o = b * KTOP + tid;
    out[o * 6 + 0] = d0;
    out[o * 6 + 1] = d1;
    out[o * 6 + 2] = d2;
    out[o * 6 + 3] = xs_w;
    out[o * 6 + 4] = z;
    out[o * 6 + 5] = ys_w;
    out[9600  + o] = score;
    out[11200 + o] = (float)cls;
    out[12800 + o] = roty;
    out[14400 + o] = alpha;
    out[16000 + o] = (score > 0.1f) ? 1.0f : 0.0f;
  }
}

extern "C" void kernel_launch(void* const* d_in, const int* in_sizes, int n_in,
                              void* d_out, int out_size, void* d_ws, size_t ws_size,
                              hipStream_t stream) {
  (void)in_sizes; (void)n_in; (void)out_size; (void)ws_size;
  const float* det_cls = (const float*)d_in[0];
  const float* det_reg = (const float*)d_in[1];
  float* s1val = (float*)d_ws;                                   // 16*400 f32
  int*   s1ind = (int*)((char*)d_ws + 16 * 400 * sizeof(float)); // 16*400 i32
  detinfer_topk_kernel<<<64, NTHR_A, SMEM_A, stream>>>(det_cls, s1val, s1ind);
  detinfer_decode_kernel<<<16, 512, 0, stream>>>(det_reg, s1val, s1ind, (float*)d_out);
}